// OptimizedMaxKSAGEConv_19894288515407
// MI455X (gfx1250) — compile-verified
//
#include <hip/hip_runtime.h>
#include <hip/hip_bf16.h>

#define D 128
#define EDGES_PER_BLOCK 1024   // 8 waves x 128 edges
#define WAVE_EDGES 128

typedef __attribute__((ext_vector_type(2))) float v2f;
typedef __attribute__((ext_vector_type(8))) float v8f;
typedef __attribute__((ext_vector_type(4))) float v4f;

// ---------------------------------------------------------------------------
// Kernel 1: zero the workspace (agg [N*D] floats followed by deg [N] floats)
// ---------------------------------------------------------------------------
__global__ void sage_zero_kernel(float* __restrict__ p, size_t n4) {
    size_t i = (size_t)blockIdx.x * blockDim.x + threadIdx.x;
    size_t stride = (size_t)gridDim.x * blockDim.x;
    v4f z = {0.f, 0.f, 0.f, 0.f};
    for (; i < n4; i += stride) {
        ((v4f*)p)[i] = z;
    }
}

// ---------------------------------------------------------------------------
// Kernel 2: edge aggregation. dst is sorted -> accumulate runs of equal dst
// in registers, flush with GLOBAL_ATOMIC_ADD_F32 at segment boundaries only.
//
// Layout: 256 threads = 8 independent waves. Block stages 1024 edge index
// pairs in LDS; wave w walks edges [w*128, w*128+128) of the block's range.
// Each lane owns 4 feature columns (float4): one global_load_b128 per lane
// per edge = one 512B wave-level load per edge (gather is L2-bandwidth-bound,
// feat fits in the 192MB L2).
// ---------------------------------------------------------------------------
__global__ void __launch_bounds__(256)
sage_aggregate_kernel(const float* __restrict__ feat,
                      const int* __restrict__ src,
                      const int* __restrict__ dst,
                      float* __restrict__ agg,
                      float* __restrict__ deg,
                      int n_edges) {
    __shared__ int s_src[EDGES_PER_BLOCK];
    __shared__ int s_dst[EDGES_PER_BLOCK];

    const int e0 = blockIdx.x * EDGES_PER_BLOCK;
    if (e0 >= n_edges) return;
    int bcount = n_edges - e0;
    if (bcount > EDGES_PER_BLOCK) bcount = EDGES_PER_BLOCK;

    for (int i = threadIdx.x; i < bcount; i += blockDim.x) {
        s_src[i] = src[e0 + i];
        s_dst[i] = dst[e0 + i];
    }
    __syncthreads();

    const int wave = threadIdx.x >> 5;     // 0..7
    const int lane = threadIdx.x & 31;
    const int col4 = lane;                 // float4 slot: columns 4*lane..4*lane+3

    const int wstart = wave * WAVE_EDGES;
    if (wstart >= bcount) return;
    int wcount = bcount - wstart;
    if (wcount > WAVE_EDGES) wcount = WAVE_EDGES;

    const v4f* __restrict__ feat4 = (const v4f*)feat;
    v4f acc = {0.f, 0.f, 0.f, 0.f};
    int cur = s_dst[wstart];               // wave-uniform
    float segcnt = 0.0f;

    for (int i = 0; i < wcount; ++i) {
        int d = s_dst[wstart + i];         // wave-uniform
        if (d != cur) {                    // wave-uniform branch
            float* base = &agg[(size_t)cur * D + col4 * 4];
            atomicAdd(base + 0, acc.x);
            atomicAdd(base + 1, acc.y);
            atomicAdd(base + 2, acc.z);
            atomicAdd(base + 3, acc.w);
            if (lane == 0) atomicAdd(&deg[cur], segcnt);
            acc = (v4f){0.f, 0.f, 0.f, 0.f};
            segcnt = 0.0f;
            cur = d;
        }
        v4f v = feat4[(size_t)s_src[wstart + i] * (D / 4) + col4]; // b128 load
        acc.x += v.x; acc.y += v.y; acc.z += v.z; acc.w += v.w;
        segcnt += 1.0f;
    }
    {
        float* base = &agg[(size_t)cur * D + col4 * 4];
        atomicAdd(base + 0, acc.x);
        atomicAdd(base + 1, acc.y);
        atomicAdd(base + 2, acc.z);
        atomicAdd(base + 3, acc.w);
        if (lane == 0) atomicAdd(&deg[cur], segcnt);
    }
}

// ---------------------------------------------------------------------------
// Kernel 3: fused out = feat @ W_self + (agg/deg) @ W_neigh + b_self
// One block (256 thr = 8 waves) per 16-node tile. Wave j owns columns
// [16j, 16j+16). K-loop in steps of 4 via V_WMMA_F32_16X16X4_F32.
//
// A frag (32-bit 16x4, ISA 7.12.2): lanes 0-15 -> M=lane, K = {0,1};
//                                   lanes 16-31 -> M=lane-16, K = {2,3}.
// B frag (4x16): N = lane&15 within VGPR; lanes 0-15 K={0,1}, 16-31 K={2,3}.
// C/D (8 VGPRs): VGPR v: lanes 0-15 -> M=v, lanes 16-31 -> M=v+8; N=lane&15.
// ---------------------------------------------------------------------------
#define LDS_PITCH 132  // 128 + 4 pad: stride-132 breaks the 64-bank conflict

__global__ void __launch_bounds__(256)
sage_gemm_kernel(const float* __restrict__ feat,
                 const float* __restrict__ W_neigh,
                 const float* __restrict__ W_self,
                 const float* __restrict__ b_self,
                 const float* __restrict__ agg,
                 const float* __restrict__ deg,
                 float* __restrict__ out,
                 int n_nodes) {
    __shared__ float sF[16 * LDS_PITCH];   // feat tile
    __shared__ float sN[16 * LDS_PITCH];   // (agg/deg) tile
    __shared__ float s_rdeg[16];

    const int t = threadIdx.x;
    const int row0 = blockIdx.x * 16;

    if (t < 16) {
        float dg = deg[row0 + t];
        s_rdeg[t] = 1.0f / fmaxf(dg, 1.0f);
    }
    __syncthreads();

    for (int idx = t; idx < 16 * D; idx += 256) {
        int r = idx >> 7;
        int c = idx & (D - 1);
        size_t g = (size_t)(row0 + r) * D + c;
        sF[r * LDS_PITCH + c] = feat[g];
        sN[r * LDS_PITCH + c] = agg[g] * s_rdeg[r];
    }
    __syncthreads();

    const int wave    = t >> 5;        // 0..7 -> N tile
    const int lane    = t & 31;
    const int halfsel = lane >> 4;     // 0 or 1
    const int l15     = lane & 15;
    const int ncol    = wave * 16 + l15;
    const int koff    = halfsel * 2;   // K offset for this half-wave
    const int m       = l15;           // A row for this lane

    v8f acc;
    {
        float bv = b_self[ncol];       // same N for all 8 accumulator regs
        for (int i = 0; i < 8; ++i) acc[i] = bv;
    }

#if __has_builtin(__builtin_amdgcn_wmma_f32_16x16x4_f32)
    for (int k0 = 0; k0 < D; k0 += 4) {
        v2f a, b;
        // self path: feat * W_self
        a.x = sF[m * LDS_PITCH + k0 + koff];
        a.y = sF[m * LDS_PITCH + k0 + koff + 1];
        b.x = W_self[(size_t)(k0 + koff) * D + ncol];
        b.y = W_self[(size_t)(k0 + koff + 1) * D + ncol];
        acc = __builtin_amdgcn_wmma_f32_16x16x4_f32(
            false, a, false, b, (short)0, acc, false, false);
        // neighbor path: (agg/deg) * W_neigh
        a.x = sN[m * LDS_PITCH + k0 + koff];
        a.y = sN[m * LDS_PITCH + k0 + koff + 1];
        b.x = W_neigh[(size_t)(k0 + koff) * D + ncol];
        b.y = W_neigh[(size_t)(k0 + koff + 1) * D + ncol];
        acc = __builtin_amdgcn_wmma_f32_16x16x4_f32(
            false, a, false, b, (short)0, acc, false, false);
    }
#else
    // scalar fallback (should not be taken on gfx1250 toolchains)
    for (int k = 0; k < D; ++k) {
        float ws = W_self[(size_t)k * D + ncol];
        float wn = W_neigh[(size_t)k * D + ncol];
        for (int i = 0; i < 8; ++i) {
            int mrow = i + halfsel * 8;
            acc[i] = fmaf(sF[mrow * LDS_PITCH + k], ws, acc[i]);
            acc[i] = fmaf(sN[mrow * LDS_PITCH + k], wn, acc[i]);
        }
    }
#endif

    for (int i = 0; i < 8; ++i) {
        int mrow = i + halfsel * 8;
        out[(size_t)(row0 + mrow) * D + ncol] = acc[i];
    }
}

// ---------------------------------------------------------------------------
// Launcher
// ---------------------------------------------------------------------------
extern "C" void kernel_launch(void* const* d_in, const int* in_sizes, int n_in,
                              void* d_out, int out_size, void* d_ws, size_t ws_size,
                              hipStream_t stream) {
    const float* feat    = (const float*)d_in[0];
    const float* W_neigh = (const float*)d_in[1];
    const float* W_self  = (const float*)d_in[2];
    const float* b_self  = (const float*)d_in[3];
    const int*   src     = (const int*)d_in[4];
    const int*   dst     = (const int*)d_in[5];

    const int n_nodes = in_sizes[0] / D;   // 100000
    const int n_edges = in_sizes[4];       // 1600000

    float* agg = (float*)d_ws;                       // n_nodes * D floats
    float* deg = agg + (size_t)n_nodes * D;          // n_nodes floats
    float* outp = (float*)d_out;

    // 1) zero agg + deg (contiguous; total is a multiple of 4 floats here)
    size_t zero_floats = (size_t)n_nodes * D + (size_t)n_nodes;
    size_t n4 = zero_floats / 4;
    sage_zero_kernel<<<2048, 256, 0, stream>>>(agg, n4);

    // 2) edge aggregation + degree counting (segment-accumulate + atomics)
    int agg_blocks = (n_edges + EDGES_PER_BLOCK - 1) / EDGES_PER_BLOCK;
    sage_aggregate_kernel<<<agg_blocks, 256, 0, stream>>>(
        feat, src, dst, agg, deg, n_edges);

    // 3) fused dual-GEMM + bias epilogue (WMMA f32 16x16x4)
    int gemm_blocks = (n_nodes + 15) / 16;   // 6250
    sage_gemm_kernel<<<gemm_blocks, 256, 0, stream>>>(
        feat, W_neigh, W_self, b_self, agg, deg, outp, n_nodes);
}